// HierarchicalAffinityLoss_14310831030886
// MI455X (gfx1250) — compile-verified
//
#include <hip/hip_runtime.h>
#include <math.h>

typedef __attribute__((ext_vector_type(2))) float v2f;
typedef __attribute__((ext_vector_type(8))) float v8f;

#define DIMK 64  // embedding dimension (K), fixed by the reference

// ---------------------------------------------------------------------------
// Kernel 1: per-row inverse L2 norm (eps-clamped) + family lookup.
// One wave (32 lanes) per row: each lane loads 2 of the 64 floats.
// ---------------------------------------------------------------------------
__global__ __launch_bounds__(256) void hal_prep_kernel(
    const float* __restrict__ emb, const int* __restrict__ labels,
    float* __restrict__ inv, int* __restrict__ fam, int n) {
  const int wid  = threadIdx.x >> 5;
  const int lane = threadIdx.x & 31;
  const int row  = blockIdx.x * 8 + wid;
  if (row >= n) return;
  float x0 = emb[row * DIMK + lane * 2 + 0];
  float x1 = emb[row * DIMK + lane * 2 + 1];
  float ss = x0 * x0 + x1 * x1;
  #pragma unroll
  for (int m = 16; m >= 1; m >>= 1) ss += __shfl_xor(ss, m, 32);
  float iv = 1.0f / fmaxf(sqrtf(ss), 1e-12f);
  if (lane == 0) {
    inv[row] = iv;
    const int table[6] = {0, 1, 2, 1, 3, 3};
    fam[row] = table[labels[row]];
  }
}

// ---------------------------------------------------------------------------
// Kernel 2: one workgroup (4 waves) per 32-row stripe.
// Each wave holds TWO 16-row A tiles (pre-normalized) and sweeps column
// tiles with stride 4. Per column tile: one set of B fragments feeds
// 32 V_WMMA_F32_16X16X4_F32 split into 4 independent accumulator chains
// (2 output tiles x even/odd K) for ILP on the matrix pipe.
// Distances are folded into running hardest-pos/neg with branchless selects.
//
// A-fragment layout (16x4 f32): lanes 0-15 -> M=lane,   K = {k0,k0+1};
//                               lanes 16-31 -> M=lane-16, K = {k0+2,k0+3}.
// B (4x16) is the symmetric layout (N in lane&15); for a Gram matrix
// B[k][n] = E[j0+n][k] == the same load pattern at row base j0.
// C/D layout: VGPR v, lane L -> element (row = v + 8*(L>>4), col = L&15).
// ---------------------------------------------------------------------------
__global__ __launch_bounds__(128) void hal_tile_kernel(
    const float* __restrict__ emb, const float* __restrict__ inv,
    const int* __restrict__ fam, float* __restrict__ per_row,
    float* __restrict__ validf, int n) {
  const int wid  = threadIdx.x >> 5;
  const int lane = threadIdx.x & 31;
  const int half = lane >> 4;   // 0 or 1
  const int l15  = lane & 15;
  const int i0   = blockIdx.x * 32;   // 32 rows per block
  const int ntiles = n >> 4;

  // Two A row-tiles for this wave, pre-normalized.
  v2f a0[16], a1[16];
  {
    const int r0 = i0 + l15;
    const int r1 = i0 + 16 + l15;
    const float iv0 = inv[r0];
    const float iv1 = inv[r1];
    const float* p0 = emb + (size_t)r0 * DIMK + half * 2;
    const float* p1 = emb + (size_t)r1 * DIMK + half * 2;
    #pragma unroll
    for (int kk = 0; kk < 16; ++kk) {
      a0[kk].x = p0[kk * 4 + 0] * iv0;
      a0[kk].y = p0[kk * 4 + 1] * iv0;
      a1[kk].x = p1[kk * 4 + 0] * iv1;
      a1[kk].y = p1[kk * 4 + 1] * iv1;
    }
  }

  // Row metadata for the 16 accumulator elements this lane owns (8 per tile).
  int famr[16];
  #pragma unroll
  for (int v = 0; v < 8; ++v) {
    famr[v]     = fam[i0 + v + 8 * half];
    famr[8 + v] = fam[i0 + 16 + v + 8 * half];
  }

  float maxpos[16], minneg[16];
  #pragma unroll
  for (int v = 0; v < 16; ++v) { maxpos[v] = -1e9f; minneg[v] = 1e9f; }

  for (int jt = wid; jt < ntiles; jt += 4) {
    const int j0 = jt * 16;
    const int colIdx = j0 + l15;

    // B fragments (raw; column normalization folded in after the WMMA).
    v2f b[16];
    {
      const float* cp = emb + (size_t)colIdx * DIMK + half * 2;
      #pragma unroll
      for (int kk = 0; kk < 16; ++kk) {
        b[kk].x = cp[kk * 4 + 0];
        b[kk].y = cp[kk * 4 + 1];
      }
    }

    // 4 independent accumulator chains (depth 8 each).
    v8f acc0e = {}, acc0o = {}, acc1e = {}, acc1o = {};
    #pragma unroll
    for (int kk = 0; kk < 8; ++kk) {
      acc0e = __builtin_amdgcn_wmma_f32_16x16x4_f32(
          false, a0[2 * kk + 0], false, b[2 * kk + 0], (short)0, acc0e, false, false);
      acc0o = __builtin_amdgcn_wmma_f32_16x16x4_f32(
          false, a0[2 * kk + 1], false, b[2 * kk + 1], (short)0, acc0o, false, false);
      acc1e = __builtin_amdgcn_wmma_f32_16x16x4_f32(
          false, a1[2 * kk + 0], false, b[2 * kk + 0], (short)0, acc1e, false, false);
      acc1o = __builtin_amdgcn_wmma_f32_16x16x4_f32(
          false, a1[2 * kk + 1], false, b[2 * kk + 1], (short)0, acc1o, false, false);
    }
    const v8f s0 = acc0e + acc0o;
    const v8f s1 = acc1e + acc1o;

    const int   famc = fam[colIdx];
    const float ivc  = inv[colIdx];
    #pragma unroll
    for (int v = 0; v < 8; ++v) {
      // Output tile 0 (rows i0 .. i0+15)
      {
        const int   rIdx = i0 + v + 8 * half;
        const float d    = 1.0f - s0[v] * ivc;
        const bool  same = (famr[v] == famc);
        const float cpos = (same && (rIdx != colIdx)) ? d : -1e9f;
        const float cneg = same ? 1e9f : d;
        maxpos[v] = fmaxf(maxpos[v], cpos);
        minneg[v] = fminf(minneg[v], cneg);
      }
      // Output tile 1 (rows i0+16 .. i0+31)
      {
        const int   rIdx = i0 + 16 + v + 8 * half;
        const float d    = 1.0f - s1[v] * ivc;
        const bool  same = (famr[8 + v] == famc);
        const float cpos = (same && (rIdx != colIdx)) ? d : -1e9f;
        const float cneg = same ? 1e9f : d;
        maxpos[8 + v] = fmaxf(maxpos[8 + v], cpos);
        minneg[8 + v] = fminf(minneg[8 + v], cneg);
      }
    }
  }

  // Reduce over the 16 columns held by each 16-lane half (xor masks 1,2,4,8
  // never cross the half boundary).
  #pragma unroll
  for (int v = 0; v < 16; ++v) {
    #pragma unroll
    for (int m = 1; m <= 8; m <<= 1) {
      maxpos[v] = fmaxf(maxpos[v], __shfl_xor(maxpos[v], m, 32));
      minneg[v] = fminf(minneg[v], __shfl_xor(minneg[v], m, 32));
    }
  }

  __shared__ float s_mp[4][32];
  __shared__ float s_mn[4][32];
  if (l15 == 0) {  // lane 0 -> local rows {0-7,16-23}, lane 16 -> {8-15,24-31}
    #pragma unroll
    for (int v = 0; v < 8; ++v) {
      s_mp[wid][v + 8 * half]      = maxpos[v];
      s_mn[wid][v + 8 * half]      = minneg[v];
      s_mp[wid][16 + v + 8 * half] = maxpos[8 + v];
      s_mn[wid][16 + v + 8 * half] = minneg[8 + v];
    }
  }
  __syncthreads();

  if (threadIdx.x < 32) {
    const int r = threadIdx.x;
    float mp = fmaxf(fmaxf(s_mp[0][r], s_mp[1][r]), fmaxf(s_mp[2][r], s_mp[3][r]));
    float mn = fminf(fminf(s_mn[0][r], s_mn[1][r]), fminf(s_mn[2][r], s_mn[3][r]));
    const bool valid = (mp > -1e8f) && (mn < 1e8f);
    per_row[i0 + r] = valid ? fmaxf(mp - mn + 0.3f, 0.0f) : 0.0f;
    validf[i0 + r]  = valid ? 1.0f : 0.0f;
  }
}

// ---------------------------------------------------------------------------
// Kernel 3: deterministic single-block final reduction.
// ---------------------------------------------------------------------------
__global__ __launch_bounds__(256) void hal_reduce_kernel(
    const float* __restrict__ per_row, const float* __restrict__ validf,
    float* __restrict__ out, int n) {
  __shared__ float ssum[256];
  __shared__ float scnt[256];
  float s = 0.0f, c = 0.0f;
  for (int i = threadIdx.x; i < n; i += 256) {
    s += per_row[i];
    c += validf[i];
  }
  ssum[threadIdx.x] = s;
  scnt[threadIdx.x] = c;
  __syncthreads();
  for (int st = 128; st > 0; st >>= 1) {
    if (threadIdx.x < st) {
      ssum[threadIdx.x] += ssum[threadIdx.x + st];
      scnt[threadIdx.x] += scnt[threadIdx.x + st];
    }
    __syncthreads();
  }
  if (threadIdx.x == 0) out[0] = ssum[0] / fmaxf(scnt[0], 1.0f);
}

// ---------------------------------------------------------------------------
// Launcher
// ---------------------------------------------------------------------------
extern "C" void kernel_launch(void* const* d_in, const int* in_sizes, int n_in,
                              void* d_out, int out_size, void* d_ws, size_t ws_size,
                              hipStream_t stream) {
  const float* emb    = (const float*)d_in[0];
  const int*   labels = (const int*)d_in[1];
  const int n = in_sizes[1];  // 8192 rows, D = 64

  float* inv     = (float*)d_ws;
  int*   fam     = (int*)(inv + n);
  float* per_row = (float*)(fam + n);
  float* validf  = per_row + n;
  float* out     = (float*)d_out;

  hal_prep_kernel<<<(n + 7) / 8, 256, 0, stream>>>(emb, labels, inv, fam, n);
  hal_tile_kernel<<<n / 32, 128, 0, stream>>>(emb, inv, fam, per_row, validf, n);
  hal_reduce_kernel<<<1, 256, 0, stream>>>(per_row, validf, out, n);
}